// GANLoss_51118700757739
// MI455X (gfx1250) — compile-verified
//
#include <hip/hip_runtime.h>

// CDNA5 / gfx1250, wave32.
typedef __attribute__((ext_vector_type(2))) float v2f;
typedef __attribute__((ext_vector_type(8))) float v8f;

#define NBLOCKS 64
#define BLOCK   256

// Kernel 1: gathered dot-product partials.
// loss_partial[b] = sum over rows handled by block b of probs[i, targets[i]] * rewards[i]
__global__ __launch_bounds__(BLOCK) void gather_dot_partial(
    const float* __restrict__ probs,
    const int*   __restrict__ targets,
    const float* __restrict__ rewards,
    float*       __restrict__ partials,
    int n_rows, int v)
{
    __shared__ float warp_part[BLOCK / 32];

    const int tid = blockIdx.x * blockDim.x + threadIdx.x;
    float acc = 0.0f;

    // One row per thread (grid exactly covers 16384 rows; loop guards generality).
    for (int i = tid; i < n_rows; i += gridDim.x * blockDim.x) {
        const int   t = targets[i];                                  // coalesced
        const float r = rewards[i];                                  // coalesced
        const float p = probs[(size_t)i * (size_t)v + (size_t)t];    // scattered gather
        acc = fmaf(p, r, acc);
    }

    // wave32 tree reduction
    #pragma unroll
    for (int off = 16; off > 0; off >>= 1)
        acc += __shfl_down(acc, off, 32);

    const int lane = threadIdx.x & 31;
    const int wid  = threadIdx.x >> 5;
    if (lane == 0) warp_part[wid] = acc;
    __syncthreads();

    if (wid == 0) {
        float w = (lane < (BLOCK / 32)) ? warp_part[lane] : 0.0f;
        #pragma unroll
        for (int off = 4; off > 0; off >>= 1)   // 8 wave-partials -> 4,2,1
            w += __shfl_down(w, off, 32);
        if (lane == 0) partials[blockIdx.x] = w;
    }
}

// Kernel 2: reduce 64 partials with one V_WMMA_F32_16X16X4_F32.
// A = ones(16x4)  =>  D[m][n] = sum_k B[k][n]  (column sums, identical rows).
// Every one of the 64 B slots (2 VGPRs x 32 lanes) holds exactly one partial,
// so  sum_n D[0][n] = sum of all 64 partials, independent of the K/N mapping.
// Row M=0 of D lives in d[0] on lanes 0..15 -> width-16 shuffle sum finishes it.
__global__ __launch_bounds__(32) void final_reduce_wmma(
    const float* __restrict__ partials,
    float*       __restrict__ out)
{
    const int lane = threadIdx.x;   // single wave32, EXEC all ones (WMMA requirement)

    v2f a; a.x = 1.0f; a.y = 1.0f;                 // A = ones 16x4
    v2f b; b.x = partials[lane];                   // 64 partials -> B operand
           b.y = partials[lane + 32];
    v8f c = {};

    // 8 args: (neg_a, A, neg_b, B, c_mod, C, reuse_a, reuse_b)
    v8f d = __builtin_amdgcn_wmma_f32_16x16x4_f32(
        /*neg_a=*/false, a, /*neg_b=*/false, b,
        /*c_mod=*/(short)0, c, /*reuse_a=*/false, /*reuse_b=*/false);

    float s = d[0];                                // D[0][n] on lanes 0..15
    #pragma unroll
    for (int off = 8; off > 0; off >>= 1)
        s += __shfl_down(s, off, 16);

    if (lane == 0) out[0] = -s;                    // loss = -dot
}

extern "C" void kernel_launch(void* const* d_in, const int* in_sizes, int n_in,
                              void* d_out, int out_size, void* d_ws, size_t ws_size,
                              hipStream_t stream) {
    const float* probs   = (const float*)d_in[0];  // (B, S, V) f32
    const int*   targets = (const int*)  d_in[1];  // (B, S)    int
    const float* rewards = (const float*)d_in[2];  // (B, S)    f32
    float*       out     = (float*)d_out;          // scalar f32
    float*       partials = (float*)d_ws;          // 64 floats of scratch

    const int n_rows = in_sizes[1];                // B*S = 16384
    const int v      = in_sizes[0] / n_rows;       // V   = 32000

    gather_dot_partial<<<NBLOCKS, BLOCK, 0, stream>>>(probs, targets, rewards,
                                                      partials, n_rows, v);
    final_reduce_wmma<<<1, 32, 0, stream>>>(partials, out);
}